// GCN_2499670966350
// MI455X (gfx1250) — compile-verified
//
#include <hip/hip_runtime.h>
#include <hip/hip_bf16.h>
#include <math.h>

#define NN   100000
#define EE   3200000
#define FIN  512
#define HID  64
#define CC   16

typedef __attribute__((ext_vector_type(2))) float v2f;
typedef __attribute__((ext_vector_type(8))) float v8f;

// ---------------------------------------------------------------------------
// Degree / normalization
// ---------------------------------------------------------------------------
__global__ void k_deg_init(float* deg) {
    int i = blockIdx.x * blockDim.x + threadIdx.x;
    if (i < NN) deg[i] = 1.0f;  // self-loop contributes weight 1.0
}

__global__ void k_deg_accum(const int* __restrict__ col,
                            const float* __restrict__ ew,
                            float* __restrict__ deg) {
    int e = blockIdx.x * blockDim.x + threadIdx.x;
    if (e < EE) atomicAdd(&deg[col[e]], ew[e]);
}

__global__ void k_dinv(const float* __restrict__ deg, float* __restrict__ dinv) {
    int i = blockIdx.x * blockDim.x + threadIdx.x;
    if (i < NN) dinv[i] = rsqrtf(deg[i]);  // deg >= 1 always (self loop)
}

__global__ void k_norm(const int* __restrict__ row, const int* __restrict__ col,
                       const float* __restrict__ ew, const float* __restrict__ dinv,
                       float* __restrict__ norm) {
    int e = blockIdx.x * blockDim.x + threadIdx.x;
    if (e < EE) norm[e] = dinv[row[e]] * ew[e] * dinv[col[e]];
}

// ---------------------------------------------------------------------------
// GEMM1: h1[N,64] = x[N,512] @ W1[64,512]^T   (fp32 WMMA 16x16x4)
// grid = 6250 blocks of 128 threads (4 waves); each wave owns one 16x16 tile.
// A frag (16x4 f32): lane(l) -> M = l&15, K = 2*(l>>4)+vgpr
// B frag (4x16 f32): lane(l) -> N = l&15, K = 2*(l>>4)+vgpr ; B[k][n] = W[n][k]
// C/D (16x16 f32):  vgpr r -> M = r + 8*(l>>4), N = l&15
// ---------------------------------------------------------------------------
__global__ void k_gemm1_wmma(const float* __restrict__ x,
                             const float* __restrict__ W1,
                             float* __restrict__ h1) {
    const int m0   = blockIdx.x * 16;
    const int wave = threadIdx.x >> 5;
    const int lane = threadIdx.x & 31;
    const int half = lane >> 4;
    const int lid  = lane & 15;
    const int n0   = wave * 16;

    const float* arow = x  + (size_t)(m0 + lid) * FIN;
    const float* brow = W1 + (size_t)(n0 + lid) * FIN;

    v8f acc0 = {}, acc1 = {}, acc2 = {}, acc3 = {};
    for (int k0 = 0; k0 < FIN; k0 += 16) {
        v2f a0 = *(const v2f*)(arow + k0 +  0 + 2 * half);
        v2f b0 = *(const v2f*)(brow + k0 +  0 + 2 * half);
        v2f a1 = *(const v2f*)(arow + k0 +  4 + 2 * half);
        v2f b1 = *(const v2f*)(brow + k0 +  4 + 2 * half);
        v2f a2 = *(const v2f*)(arow + k0 +  8 + 2 * half);
        v2f b2 = *(const v2f*)(brow + k0 +  8 + 2 * half);
        v2f a3 = *(const v2f*)(arow + k0 + 12 + 2 * half);
        v2f b3 = *(const v2f*)(brow + k0 + 12 + 2 * half);
        acc0 = __builtin_amdgcn_wmma_f32_16x16x4_f32(false, a0, false, b0, (short)0, acc0, false, false);
        acc1 = __builtin_amdgcn_wmma_f32_16x16x4_f32(false, a1, false, b1, (short)0, acc1, false, false);
        acc2 = __builtin_amdgcn_wmma_f32_16x16x4_f32(false, a2, false, b2, (short)0, acc2, false, false);
        acc3 = __builtin_amdgcn_wmma_f32_16x16x4_f32(false, a3, false, b3, (short)0, acc3, false, false);
    }
    v8f acc = (acc0 + acc1) + (acc2 + acc3);

    float* orow = h1 + (size_t)(m0 + 8 * half) * HID + n0 + lid;
#pragma unroll
    for (int r = 0; r < 8; ++r) orow[(size_t)r * HID] = acc[r];
}

// ---------------------------------------------------------------------------
// GEMM2: h2[N,16] = hr[N,64] @ W2[16,64]^T   (fp32 WMMA 16x16x4)
// 8 waves/block, one 16x16 tile per wave, 6250 tiles total.
// ---------------------------------------------------------------------------
__global__ void k_gemm2_wmma(const float* __restrict__ hr,
                             const float* __restrict__ W2,
                             float* __restrict__ h2) {
    const int mt = blockIdx.x * (blockDim.x >> 5) + (threadIdx.x >> 5);
    if (mt >= NN / 16) return;  // wave-uniform exit: EXEC stays all-ones for WMMA
    const int m0   = mt * 16;
    const int lane = threadIdx.x & 31;
    const int half = lane >> 4;
    const int lid  = lane & 15;

    const float* arow = hr + (size_t)(m0 + lid) * HID;
    const float* brow = W2 + (size_t)lid * HID;  // n0 = 0, C = 16

    v8f acc0 = {}, acc1 = {}, acc2 = {}, acc3 = {};
    for (int k0 = 0; k0 < HID; k0 += 16) {
        v2f a0 = *(const v2f*)(arow + k0 +  0 + 2 * half);
        v2f b0 = *(const v2f*)(brow + k0 +  0 + 2 * half);
        v2f a1 = *(const v2f*)(arow + k0 +  4 + 2 * half);
        v2f b1 = *(const v2f*)(brow + k0 +  4 + 2 * half);
        v2f a2 = *(const v2f*)(arow + k0 +  8 + 2 * half);
        v2f b2 = *(const v2f*)(brow + k0 +  8 + 2 * half);
        v2f a3 = *(const v2f*)(arow + k0 + 12 + 2 * half);
        v2f b3 = *(const v2f*)(brow + k0 + 12 + 2 * half);
        acc0 = __builtin_amdgcn_wmma_f32_16x16x4_f32(false, a0, false, b0, (short)0, acc0, false, false);
        acc1 = __builtin_amdgcn_wmma_f32_16x16x4_f32(false, a1, false, b1, (short)0, acc1, false, false);
        acc2 = __builtin_amdgcn_wmma_f32_16x16x4_f32(false, a2, false, b2, (short)0, acc2, false, false);
        acc3 = __builtin_amdgcn_wmma_f32_16x16x4_f32(false, a3, false, b3, (short)0, acc3, false, false);
    }
    v8f acc = (acc0 + acc1) + (acc2 + acc3);

    float* orow = h2 + (size_t)(m0 + 8 * half) * CC + lid;
#pragma unroll
    for (int r = 0; r < 8; ++r) orow[(size_t)r * CC] = acc[r];
}

// ---------------------------------------------------------------------------
// Aggregation: self-loop init, then edge scatter-add (F = 64 or 16).
// h1/out fit in the 192MB L2, so the random gather + f32 atomics stay on-chip.
// ---------------------------------------------------------------------------
__global__ void k_selfloop_init(const float* __restrict__ h,
                                const float* __restrict__ dinv,
                                float* __restrict__ out, int F) {
    int t = blockIdx.x * blockDim.x + threadIdx.x;
    if (t >= NN * F) return;
    int i = t / F;
    float d = dinv[i];
    out[t] = h[t] * d * d;
}

// One edge handled by F/4 threads; each thread moves a float4.
__global__ void k_edge_agg(const float* __restrict__ h,
                           const int* __restrict__ row, const int* __restrict__ col,
                           const float* __restrict__ norm,
                           float* __restrict__ out, int F) {
    const int tpe = F >> 2;  // threads per edge
    long long t = (long long)blockIdx.x * blockDim.x + threadIdx.x;
    if (t >= (long long)EE * tpe) return;
    int e = (int)(t / tpe);
    int f = ((int)(t % tpe)) << 2;
    int r = row[e], c = col[e];
    float w = norm[e];
    float4 v = *(const float4*)(h + (size_t)r * F + f);
    float* dst = out + (size_t)c * F + f;
    atomicAdd(dst + 0, v.x * w);
    atomicAdd(dst + 1, v.y * w);
    atomicAdd(dst + 2, v.z * w);
    atomicAdd(dst + 3, v.w * w);
}

__global__ void k_bias_relu(float* __restrict__ h, const float* __restrict__ b) {
    int t = blockIdx.x * blockDim.x + threadIdx.x;
    if (t >= NN * HID) return;
    h[t] = fmaxf(h[t] + b[t % HID], 0.0f);
}

__global__ void k_log_softmax(float* __restrict__ out, const float* __restrict__ b2) {
    int i = blockIdx.x * blockDim.x + threadIdx.x;
    if (i >= NN) return;
    float v[CC];
    float m = -INFINITY;
#pragma unroll
    for (int f = 0; f < CC; ++f) {
        v[f] = out[(size_t)i * CC + f] + b2[f];
        m = fmaxf(m, v[f]);
    }
    float s = 0.0f;
#pragma unroll
    for (int f = 0; f < CC; ++f) s += expf(v[f] - m);
    float ls = m + logf(s);
#pragma unroll
    for (int f = 0; f < CC; ++f) out[(size_t)i * CC + f] = v[f] - ls;
}

// ---------------------------------------------------------------------------
extern "C" void kernel_launch(void* const* d_in, const int* in_sizes, int n_in,
                              void* d_out, int out_size, void* d_ws, size_t ws_size,
                              hipStream_t stream) {
    (void)in_sizes; (void)n_in; (void)out_size; (void)ws_size;

    const float* x   = (const float*)d_in[0];
    const int*   ei  = (const int*)d_in[1];
    const float* ea  = (const float*)d_in[2];
    const float* W1  = (const float*)d_in[3];
    const float* b1  = (const float*)d_in[4];
    const float* W2  = (const float*)d_in[5];
    const float* b2  = (const float*)d_in[6];
    float* out = (float*)d_out;

    const int* row = ei;        // edge_index[0]
    const int* col = ei + EE;   // edge_index[1]

    // Carve workspace (~71 MB)
    char* ws = (char*)d_ws;
    size_t off = 0;
    auto carve = [&](size_t bytes) -> char* {
        char* p = ws + off;
        off = (off + bytes + 255) & ~(size_t)255;
        return p;
    };
    float* deg  = (float*)carve((size_t)NN * 4);
    float* dinv = (float*)carve((size_t)NN * 4);
    float* norm = (float*)carve((size_t)EE * 4);
    float* h1   = (float*)carve((size_t)NN * HID * 4);
    float* agg1 = (float*)carve((size_t)NN * HID * 4);
    float* h2   = (float*)carve((size_t)NN * CC * 4);

    const int B = 256;

    // 1) symmetric normalization coefficients
    k_deg_init<<<(NN + B - 1) / B, B, 0, stream>>>(deg);
    k_deg_accum<<<(EE + B - 1) / B, B, 0, stream>>>(col, ea, deg);
    k_dinv<<<(NN + B - 1) / B, B, 0, stream>>>(deg, dinv);
    k_norm<<<(EE + B - 1) / B, B, 0, stream>>>(row, col, ea, dinv, norm);

    // 2) layer 1: GEMM (fp32 WMMA) -> aggregate -> bias+ReLU
    k_gemm1_wmma<<<NN / 16, 128, 0, stream>>>(x, W1, h1);
    k_selfloop_init<<<(NN * HID + B - 1) / B, B, 0, stream>>>(h1, dinv, agg1, HID);
    {
        long long threads = (long long)EE * (HID / 4);
        k_edge_agg<<<(unsigned)((threads + B - 1) / B), B, 0, stream>>>(h1, row, col, norm, agg1, HID);
    }
    k_bias_relu<<<(NN * HID + B - 1) / B, B, 0, stream>>>(agg1, b1);

    // 3) layer 2: GEMM (fp32 WMMA) -> aggregate into d_out -> log_softmax
    k_gemm2_wmma<<<(NN / 16 + 7) / 8, 256, 0, stream>>>(agg1, W2, h2);
    k_selfloop_init<<<(NN * CC + B - 1) / B, B, 0, stream>>>(h2, dinv, out, CC);
    {
        long long threads = (long long)EE * (CC / 4);
        k_edge_agg<<<(unsigned)((threads + B - 1) / B), B, 0, stream>>>(h2, row, col, norm, out, CC);
    }
    k_log_softmax<<<(NN + B - 1) / B, B, 0, stream>>>(out, b2);
}